// GAT_19404662243879
// MI455X (gfx1250) — compile-verified
//
#include <hip/hip_runtime.h>
#include <math.h>

typedef float v2f __attribute__((ext_vector_type(2)));
typedef float v4f __attribute__((ext_vector_type(4)));
typedef float v8f __attribute__((ext_vector_type(8)));

#define D 128
#define XS_STRIDE 132   // 16x128 fp32 x-tile in LDS, padded: conflict-free ds_load_b64 frags
#define NEG_SLOPE 0.01f

// Non-returning device-scope fp32 atomic add (global_atomic_add_f32, STOREcnt-tracked).
__device__ __forceinline__ void atomic_add_f32_dev(float* p, float v) {
    asm volatile("global_atomic_add_f32 %0, %1, off scope:SCOPE_DEV"
                 :: "v"(p), "v"(v)
                 : "memory");
}

// ---------------------------------------------------------------------------
// Kernel 0: zero d_out (agg accumulator) and denom.
// ---------------------------------------------------------------------------
__global__ __launch_bounds__(256) void zero_kernel(float* __restrict__ out,
                                                   float* __restrict__ denom,
                                                   int n_out4, int n_den4) {
    int i = blockIdx.x * 256 + threadIdx.x;
    v4f z = {};
    if (i < n_out4) {
        ((v4f*)out)[i] = z;
    } else if (i < n_out4 + n_den4) {
        ((v4f*)denom)[i - n_out4] = z;
    }
}

// ---------------------------------------------------------------------------
// Kernel 1: h = x @ W^T + b via V_WMMA_F32_16X16X4_F32.
// Block = 256 threads = 8 waves. Wave w owns output columns [16w,16w+16).
// Block covers 8 node-tiles of 16 (128 nodes). B-frags (W) preloaded to regs.
// ---------------------------------------------------------------------------
__global__ __launch_bounds__(256) void gemm_wmma_kernel(const float* __restrict__ x,
                                                        const float* __restrict__ W,
                                                        const float* __restrict__ b,
                                                        float* __restrict__ h,
                                                        int N) {
    __shared__ float xs[16 * XS_STRIDE];

    const int wave = threadIdx.x >> 5;   // 0..7
    const int lane = threadIdx.x & 31;
    const int kh   = lane >> 4;          // half of wave: selects K pair
    const int lrow = lane & 15;
    const int obase = wave * 16;

    // Preload B fragments: B[k][n] = W^T[k][o] = W[obase+lrow][k], all K=128.
    // lane l holds K = 4s + 2*kh + {0,1}  -> contiguous float2 per step.
    v2f bfrag[32];
#pragma unroll
    for (int s = 0; s < 32; ++s) {
        bfrag[s] = *(const v2f*)(W + (obase + lrow) * D + (s * 4 + kh * 2));
    }
    const float bias = b[obase + lrow];

    const int nodeBlock = blockIdx.x * 128;
    for (int t = 0; t < 8; ++t) {
        const int node0 = nodeBlock + t * 16;
        if (node0 >= N) break;          // N % 16 == 0: whole tiles only

        __syncthreads();                // previous tile's readers done
        // Cooperative coalesced load of x[node0..node0+15][0..127] into LDS.
        {
            int f = threadIdx.x;        // float4 index, 512 total over 2 rounds
#pragma unroll
            for (int r = 0; r < 2; ++r, f += 256) {
                int row = f >> 5, c4 = f & 31;
                v4f v = *(const v4f*)(x + (size_t)(node0 + row) * D + c4 * 4);
                *(v4f*)(&xs[row * XS_STRIDE + c4 * 4]) = v;
            }
        }
        __syncthreads();

        v8f acc = {};
#pragma unroll
        for (int s = 0; s < 32; ++s) {
            // A-frag: lane l holds x[node0 + l%16][4s + 2*(l/16) + {0,1}]
            v2f afrag = *(const v2f*)(&xs[lrow * XS_STRIDE + s * 4 + kh * 2]);
            acc = __builtin_amdgcn_wmma_f32_16x16x4_f32(
                      /*neg_a=*/false, afrag, /*neg_b=*/false, bfrag[s],
                      /*c_mod=*/(short)0, acc, /*reuse_a=*/false, /*reuse_b=*/false);
        }

        // C layout: VGPR r, lanes 0-15 -> row r; lanes 16-31 -> row 8+r.
#pragma unroll
        for (int r = 0; r < 8; ++r) {
            h[(size_t)(node0 + r + 8 * kh) * D + obase + lrow] = acc[r] + bias;
        }
    }
}

// ---------------------------------------------------------------------------
// Kernel 2: per-edge attention logits -> e = exp(leakyrelu(score)),
// denom[src] += e.  8 lanes per edge, float4 loads, shfl_xor reduce.
// Edges [0,E) from edge_index; edges [E,E+N) are self-loops.
// ---------------------------------------------------------------------------
__global__ __launch_bounds__(256) void score_kernel(const int* __restrict__ ei,
                                                    const float* __restrict__ h,
                                                    const float* __restrict__ Watt,
                                                    const float* __restrict__ batt,
                                                    float* __restrict__ ebuf,
                                                    float* __restrict__ denom,
                                                    int E, int M) {
    const int tid  = blockIdx.x * 256 + threadIdx.x;
    const int edge = tid >> 3;
    const int g    = tid & 7;
    if (edge >= M) return;

    int src, dst;
    if (edge < E) { src = ei[edge]; dst = ei[E + edge]; }
    else          { src = dst = edge - E; }

    const float* hs = h + (size_t)src * D;
    const float* hd = h + (size_t)dst * D;

    float p = 0.f;
#pragma unroll
    for (int j = 0; j < 4; ++j) {
        const int c = g * 4 + j * 32;
        v4f a  = *(const v4f*)(hs + c);
        v4f wi = *(const v4f*)(Watt + c);
        v4f d  = *(const v4f*)(hd + c);
        v4f wj = *(const v4f*)(Watt + D + c);
        p += a.x * wi.x + a.y * wi.y + a.z * wi.z + a.w * wi.w;
        p += d.x * wj.x + d.y * wj.y + d.z * wj.z + d.w * wj.w;
    }
    // reduce across the 8-lane group
    p += __shfl_xor(p, 1, 32);
    p += __shfl_xor(p, 2, 32);
    p += __shfl_xor(p, 4, 32);

    if (g == 0) {
        float s = p + batt[0];
        s = (s >= 0.f) ? s : NEG_SLOPE * s;
        float el = expf(s);
        ebuf[edge] = el;
        atomic_add_f32_dev(denom + src, el);
    }
}

// ---------------------------------------------------------------------------
// Kernel 3: agg[src] += (e/denom[src]) * h[dst].  8 lanes per edge,
// 16 global_atomic_add_f32 per lane.
// ---------------------------------------------------------------------------
__global__ __launch_bounds__(256) void agg_kernel(const int* __restrict__ ei,
                                                  const float* __restrict__ h,
                                                  const float* __restrict__ ebuf,
                                                  const float* __restrict__ denom,
                                                  float* __restrict__ out,
                                                  int E, int M) {
    const int tid  = blockIdx.x * 256 + threadIdx.x;
    const int edge = tid >> 3;
    const int g    = tid & 7;
    if (edge >= M) return;

    int src, dst;
    if (edge < E) { src = ei[edge]; dst = ei[E + edge]; }
    else          { src = dst = edge - E; }

    const float alpha = ebuf[edge] / denom[src];
    const float* hd = h + (size_t)dst * D;
    float* op = out + (size_t)src * D;

#pragma unroll
    for (int j = 0; j < 4; ++j) {
        const int c = g * 4 + j * 32;
        v4f v = *(const v4f*)(hd + c);
        atomic_add_f32_dev(op + c + 0, alpha * v.x);
        atomic_add_f32_dev(op + c + 1, alpha * v.y);
        atomic_add_f32_dev(op + c + 2, alpha * v.z);
        atomic_add_f32_dev(op + c + 3, alpha * v.w);
    }
}

// ---------------------------------------------------------------------------
// Kernel 4: out = elu(out), in place.
// ---------------------------------------------------------------------------
__global__ __launch_bounds__(256) void elu_kernel(float* __restrict__ out, int n4) {
    int i = blockIdx.x * 256 + threadIdx.x;
    if (i >= n4) return;
    v4f v = ((const v4f*)out)[i];
    v.x = (v.x > 0.f) ? v.x : (expf(v.x) - 1.f);
    v.y = (v.y > 0.f) ? v.y : (expf(v.y) - 1.f);
    v.z = (v.z > 0.f) ? v.z : (expf(v.z) - 1.f);
    v.w = (v.w > 0.f) ? v.w : (expf(v.w) - 1.f);
    ((v4f*)out)[i] = v;
}

// ---------------------------------------------------------------------------
extern "C" void kernel_launch(void* const* d_in, const int* in_sizes, int n_in,
                              void* d_out, int out_size, void* d_ws, size_t ws_size,
                              hipStream_t stream) {
    const float* x    = (const float*)d_in[0];
    const int*   ei   = (const int*)  d_in[1];
    const float* Wlin = (const float*)d_in[2];
    const float* blin = (const float*)d_in[3];
    const float* Watt = (const float*)d_in[4];
    const float* batt = (const float*)d_in[5];

    const int N = in_sizes[0] / D;      // 100000
    const int E = in_sizes[1] / 2;      // 1600000
    const int M = E + N;                // edges incl. self-loops

    float* out = (float*)d_out;

    // workspace layout (fp32 elements): h[N*D] | ebuf[M] | denom[N]
    float* h     = (float*)d_ws;
    float* ebuf  = h + (size_t)N * D;
    float* denom = ebuf + (((size_t)M + 3) & ~(size_t)3);   // keep 16B alignment

    const int n_out4 = (N * D) / 4;
    const int n_den4 = (N + 3) / 4;

    // 0) zero accumulators
    {
        int total = n_out4 + n_den4;
        zero_kernel<<<(total + 255) / 256, 256, 0, stream>>>(out, denom, n_out4, n_den4);
    }
    // 1) h = x @ W^T + b  (WMMA f32)
    {
        int blocks = (N + 127) / 128;
        gemm_wmma_kernel<<<blocks, 256, 0, stream>>>(x, Wlin, blin, h, N);
    }
    // 2) edge logits + softmax denominator
    {
        long long threads = (long long)M * 8;
        int blocks = (int)((threads + 255) / 256);
        score_kernel<<<blocks, 256, 0, stream>>>(ei, h, Watt, batt, ebuf, denom, E, M);
    }
    // 3) weighted aggregation into out
    {
        long long threads = (long long)M * 8;
        int blocks = (int)((threads + 255) / 256);
        agg_kernel<<<blocks, 256, 0, stream>>>(ei, h, ebuf, denom, out, E, M);
    }
    // 4) elu in place
    {
        elu_kernel<<<(n_out4 + 255) / 256, 256, 0, stream>>>(out, n_out4);
    }
}